// GNNModel_80951543595551
// MI455X (gfx1250) — compile-verified
//
#include <hip/hip_runtime.h>

typedef __attribute__((ext_vector_type(2))) float v2f;
typedef __attribute__((ext_vector_type(8))) float v8f;

#define GNN_N 100000
#define GNN_E 1600000
#define IN_F 128
#define HID 128
#define HID2 256
#define OUT_F 64

// -------------------- CDNA5 async global->LDS copy helpers --------------------
// One 16-byte chunk per lane, per issue. Tracked with ASYNCcnt.

__device__ __forceinline__ void async_copy_b128(unsigned lds_off, const void* gaddr) {
    unsigned long long ga = (unsigned long long)(uintptr_t)gaddr;
    asm volatile("global_load_async_to_lds_b128 %0, %1, off"
                 :: "v"(lds_off), "v"(ga) : "memory");
}

__device__ __forceinline__ void async_wait0() {
    asm volatile("s_wait_asynccnt 0x0" ::: "memory");
}

// -------------------- degree / norm --------------------

__global__ void degree_kernel(const int* __restrict__ src, const int* __restrict__ dst,
                              float* __restrict__ deg_out, float* __restrict__ deg_in, int E) {
    int e = blockIdx.x * blockDim.x + threadIdx.x;
    if (e < E) {
        atomicAdd(&deg_out[src[e]], 1.0f);
        atomicAdd(&deg_in[dst[e]], 1.0f);
    }
}

__global__ void norm_kernel(const float* __restrict__ deg_out, const float* __restrict__ deg_in,
                            float* __restrict__ ns, float* __restrict__ nd, int n) {
    int i = blockIdx.x * blockDim.x + threadIdx.x;
    if (i < n) {
        ns[i] = rsqrtf(fmaxf(deg_out[i], 1.0f));
        nd[i] = rsqrtf(fmaxf(deg_in[i], 1.0f));
    }
}

// -------------------- edge scatter: agg[dst] += x[src] * ns[src] --------------------
// One wave (32 lanes) per edge; each lane moves 4 contiguous floats (float4 load,
// 4 global_atomic_add_f32).

__global__ void scatter_kernel(const float* __restrict__ x, const float* __restrict__ ns,
                               const int* __restrict__ src, const int* __restrict__ dst,
                               float* __restrict__ agg, int E) {
    long gid = (long)blockIdx.x * blockDim.x + threadIdx.x;
    int edge = (int)(gid >> 5);
    int lane = (int)(gid & 31);
    if (edge >= E) return;
    int s = src[edge];
    int d = dst[edge];
    float scale = ns[s];
    const float4* xr = (const float4*)(x + (size_t)s * 128);
    float4 v = xr[lane];
    float* ar = agg + (size_t)d * 128 + lane * 4;
    atomicAdd(ar + 0, v.x * scale);
    atomicAdd(ar + 1, v.y * scale);
    atomicAdd(ar + 2, v.z * scale);
    atomicAdd(ar + 3, v.w * scale);
}

// -------------------- fused (diag(nd) * A) @ W + bias via WMMA fp32 --------------------
// diag(nd)*A*W == diag(nd)*(A*W), so A rows are async-copied raw into LDS and the
// per-row nd scale is applied to the WMMA accumulator at store time.
// One workgroup (8 waves) per 16-row block; each wave owns 16x16 output tiles,
// chaining v_wmma_f32_16x16x4_f32 over K.

#define ALDS_STR 132  // 128 + 4 pad floats; 528 B/row keeps 16B alignment + bank spread

__global__ void gemm_nd_kernel(const float* __restrict__ A, const float* __restrict__ nd,
                               const float* __restrict__ W, const float* __restrict__ bias,
                               float* __restrict__ out, int K, int C) {
    __shared__ float Alds[16 * ALDS_STR];
    const int tid = threadIdx.x;
    const size_t rowbase = (size_t)blockIdx.x * 16;

    // Async-stage raw A block (16 x K) into LDS, 16B chunks per lane.
    const int chunks = 16 * (K / 4);
    for (int i = tid; i < chunks; i += 256) {
        int r = i / (K / 4), c4 = (i % (K / 4)) * 4;
        unsigned lds_off = (unsigned)(uintptr_t)&Alds[r * ALDS_STR + c4];
        async_copy_b128(lds_off, A + (rowbase + r) * K + c4);
    }
    async_wait0();
    __syncthreads();

    const int lane = tid & 31;
    const int wv = tid >> 5;
    const int half = lane >> 4;   // 0: K pair {k0,k0+1}; 1: {k0+2,k0+3}
    const int mn = lane & 15;     // A: row index M; B: col index N

    for (int t = wv; t * 16 < C; t += 8) {
        const int colbase = t * 16;
        v8f acc = {};
        for (int k0 = 0; k0 < K; k0 += 4) {
            const int ka = k0 + 2 * half;
            v2f a, b;
            a.x = Alds[mn * ALDS_STR + ka];
            a.y = Alds[mn * ALDS_STR + ka + 1];
            b.x = W[(size_t)ka * C + colbase + mn];
            b.y = W[(size_t)(ka + 1) * C + colbase + mn];
            acc = __builtin_amdgcn_wmma_f32_16x16x4_f32(false, a, false, b,
                                                        (short)0, acc, false, false);
        }
        const float bv = bias[colbase + mn];
        #pragma unroll
        for (int r = 0; r < 8; ++r) {
            const int m = r + 8 * half;
            out[(rowbase + m) * C + colbase + mn] = nd[rowbase + m] * acc[r] + bv;
        }
    }
}

// -------------------- fused MLP + proj --------------------
// per 16-row block: T = relu(h2 @ W3 + b3)  [16 x 256]  (stage in LDS)
//                   O = T @ W4 + b4         [16 x 64]   (stage in LDS)
//                   out = max(eps, |O / rownorm(O)|)

#define TLDS_STR 260
#define OLDS_STR 68

__global__ void mlp_proj_kernel(const float* __restrict__ h2,
                                const float* __restrict__ W3, const float* __restrict__ b3,
                                const float* __restrict__ W4, const float* __restrict__ b4,
                                float* __restrict__ out) {
    __shared__ float Alds[16 * ALDS_STR];
    __shared__ float Tlds[16 * TLDS_STR];
    __shared__ float Olds[16 * OLDS_STR];
    __shared__ float rninv[16];

    const int tid = threadIdx.x;
    const size_t rowbase = (size_t)blockIdx.x * 16;

    // Async-stage h2 block (16 x 128) into LDS.
    for (int i = tid; i < 16 * (HID / 4); i += 256) {
        int r = i / (HID / 4), c4 = (i % (HID / 4)) * 4;
        unsigned lds_off = (unsigned)(uintptr_t)&Alds[r * ALDS_STR + c4];
        async_copy_b128(lds_off, h2 + (rowbase + r) * HID + c4);
    }
    async_wait0();
    __syncthreads();

    const int lane = tid & 31;
    const int wv = tid >> 5;
    const int half = lane >> 4;
    const int mn = lane & 15;

    // Stage 1: [16x128] @ [128x256] + b3, ReLU -> Tlds. 16 col tiles over 8 waves.
    for (int t = wv; t < HID2 / 16; t += 8) {
        const int colbase = t * 16;
        v8f acc = {};
        for (int k0 = 0; k0 < HID; k0 += 4) {
            const int ka = k0 + 2 * half;
            v2f a, b;
            a.x = Alds[mn * ALDS_STR + ka];
            a.y = Alds[mn * ALDS_STR + ka + 1];
            b.x = W3[(size_t)ka * HID2 + colbase + mn];
            b.y = W3[(size_t)(ka + 1) * HID2 + colbase + mn];
            acc = __builtin_amdgcn_wmma_f32_16x16x4_f32(false, a, false, b,
                                                        (short)0, acc, false, false);
        }
        const float bv = b3[colbase + mn];
        #pragma unroll
        for (int r = 0; r < 8; ++r) {
            const int m = r + 8 * half;
            Tlds[m * TLDS_STR + colbase + mn] = fmaxf(acc[r] + bv, 0.0f);
        }
    }
    __syncthreads();

    // Stage 2: [16x256] @ [256x64] + b4 -> Olds. 4 col tiles on waves 0..3.
    if (wv < 4) {
        const int colbase = wv * 16;
        v8f acc = {};
        for (int k0 = 0; k0 < HID2; k0 += 4) {
            const int ka = k0 + 2 * half;
            v2f a, b;
            a.x = Tlds[mn * TLDS_STR + ka];
            a.y = Tlds[mn * TLDS_STR + ka + 1];
            b.x = W4[(size_t)ka * OUT_F + colbase + mn];
            b.y = W4[(size_t)(ka + 1) * OUT_F + colbase + mn];
            acc = __builtin_amdgcn_wmma_f32_16x16x4_f32(false, a, false, b,
                                                        (short)0, acc, false, false);
        }
        const float bv = b4[colbase + mn];
        #pragma unroll
        for (int r = 0; r < 8; ++r) {
            const int m = r + 8 * half;
            Olds[m * OLDS_STR + colbase + mn] = acc[r] + bv;
        }
    }
    __syncthreads();

    // Row L2 norms.
    if (tid < 16) {
        float s = 0.0f;
        for (int c = 0; c < OUT_F; ++c) {
            float v = Olds[tid * OLDS_STR + c];
            s += v * v;
        }
        rninv[tid] = rsqrtf(fmaxf(s, 1e-30f));
    }
    __syncthreads();

    // proj: max(eps, |x / norm|)
    for (int i = tid; i < 16 * OUT_F; i += 256) {
        int r = i / OUT_F, c = i % OUT_F;
        float v = Olds[r * OLDS_STR + c] * rninv[r];
        out[(rowbase + r) * OUT_F + c] = fmaxf(1e-6f, fabsf(v));
    }
}

// -------------------- host launch --------------------

extern "C" void kernel_launch(void* const* d_in, const int* in_sizes, int n_in,
                              void* d_out, int out_size, void* d_ws, size_t ws_size,
                              hipStream_t stream) {
    const float* feat = (const float*)d_in[0];
    const int* src    = (const int*)d_in[1];
    const int* dst    = (const int*)d_in[2];
    const float* W1   = (const float*)d_in[3];
    const float* b1   = (const float*)d_in[4];
    const float* W2   = (const float*)d_in[5];
    const float* b2   = (const float*)d_in[6];
    const float* W3   = (const float*)d_in[7];
    const float* b3   = (const float*)d_in[8];
    const float* W4   = (const float*)d_in[9];
    const float* b4   = (const float*)d_in[10];
    float* outp = (float*)d_out;

    const int N = GNN_N;
    const int E = GNN_E;

    float* ws = (float*)d_ws;
    float* deg_out = ws;                 // N
    float* deg_in  = ws + (size_t)N;     // N
    float* ns      = ws + (size_t)2 * N; // N
    float* nd      = ws + (size_t)3 * N; // N
    float* agg     = ws + (size_t)4 * N;                    // N*128
    float* h       = ws + (size_t)4 * N + (size_t)N * 128;  // N*128

    hipMemsetAsync(deg_out, 0, (size_t)2 * N * sizeof(float), stream);
    hipMemsetAsync(agg, 0, (size_t)N * 128 * sizeof(float), stream);

    degree_kernel<<<(E + 255) / 256, 256, 0, stream>>>(src, dst, deg_out, deg_in, E);
    norm_kernel<<<(N + 255) / 256, 256, 0, stream>>>(deg_out, deg_in, ns, nd, N);

    const int scatterBlocks = (int)(((long)E * 32 + 255) / 256);
    const int rowBlocks = N / 16;  // 100000 / 16 = 6250, exact

    // Layer 1
    scatter_kernel<<<scatterBlocks, 256, 0, stream>>>(feat, ns, src, dst, agg, E);
    gemm_nd_kernel<<<rowBlocks, 256, 0, stream>>>(agg, nd, W1, b1, h, IN_F, HID);

    // Layer 2
    hipMemsetAsync(agg, 0, (size_t)N * 128 * sizeof(float), stream);
    scatter_kernel<<<scatterBlocks, 256, 0, stream>>>(h, ns, src, dst, agg, E);
    gemm_nd_kernel<<<rowBlocks, 256, 0, stream>>>(agg, nd, W2, b2, h, HID, HID);

    // MLP + proj
    mlp_proj_kernel<<<rowBlocks, 256, 0, stream>>>(h, W3, b3, W4, b4, outp);
}